// GraphAttentionLayer_9749575762612
// MI455X (gfx1250) — compile-verified
//
#include <hip/hip_runtime.h>
#include <math.h>
#include <stdint.h>

// -------- problem constants (from reference setup_inputs) --------
#define BB   16
#define NN   200
#define DD   128
#define HH   8
#define NPH  16              // n = D/H
#define ROWS (BB * NN)       // 3200 flattened (b,i) rows
#define EROWS ((size_t)BB * NN * NN) // 640000 flattened (b,i,j) rows
#define NEG_SLOPE 0.2f

// padded LDS row stride for 16x128 fp32 A-tiles: 132 % 64 == 4 banks  ->
// lanes 0..15 reading row-strided float2 hit distinct banks (4*l15 + k)
#define APAD 132

typedef __attribute__((ext_vector_type(2))) float v2f;
typedef __attribute__((ext_vector_type(8))) float v8f;

// ---- CDNA5 async memory->LDS copy (ASYNCcnt-tracked), 16B per lane ----
__device__ __forceinline__ void async_copy_b128(const float* gsrc, float* lds_dst) {
  const unsigned lds_addr = (unsigned)(uintptr_t)lds_dst;   // low 32b = LDS byte addr
  asm volatile("global_load_async_to_lds_b128 %0, %1, off"
               :: "v"(lds_addr), "v"(gsrc) : "memory");
}
__device__ __forceinline__ void async_wait0() {
  asm volatile("s_wait_asynccnt 0" ::: "memory");
}

// =====================================================================
// Kernel 1: fold attention vector `a` into the weight matrices.
//   u1[d][h] = sum_n W [d][16h+n] * a[n]
//   u2[d][h] = sum_n W [d][16h+n] * a[16+n]
//   v  [d][h] = sum_n We[d][16h+n] * a[32+n]   (v padded to 16 cols, cols 8..15 = 0)
// =====================================================================
__global__ void gat_prep_kernel(const float* __restrict__ W,
                                const float* __restrict__ We,
                                const float* __restrict__ a,
                                float* __restrict__ u1,
                                float* __restrict__ u2,
                                float* __restrict__ vpad) {
  const int d = threadIdx.x;            // 0..127
  for (int h = 0; h < HH; ++h) {
    float su1 = 0.f, su2 = 0.f, sv = 0.f;
    #pragma unroll
    for (int n = 0; n < NPH; ++n) {
      const float w  = W [d * DD + h * NPH + n];
      const float we = We[d * DD + h * NPH + n];
      su1 += w  * a[n];
      su2 += w  * a[NPH + n];
      sv  += we * a[2 * NPH + n];
    }
    u1[d * HH + h] = su1;
    u2[d * HH + h] = su2;
    vpad[d * 16 + h]     = sv;
    vpad[d * 16 + 8 + h] = 0.f;
  }
}

// =====================================================================
// Kernel 2: g = h @ W via V_WMMA_F32_16X16X4_F32.
// Block = 256 thr (8 waves). Block owns one 16-row A tile (async-staged
// once into LDS, shared by all 8 waves); wave w computes cols [16w,16w+16).
// K loop: 32 steps of k=4.  W (64 KB) stays L2-hot, read directly.
// =====================================================================
__global__ void gat_gemm_g_kernel(const float* __restrict__ hmat,
                                  const float* __restrict__ W,
                                  float* __restrict__ g) {
  __shared__ __align__(16) float As[16 * APAD];
  const int tid  = threadIdx.x;
  const int row0 = blockIdx.x * 16;

  // async-stage 16x128 A tile (512 float4 / 256 thr -> 2 issues each)
  #pragma unroll
  for (int fi = tid; fi < 512; fi += 256) {
    const int r = fi >> 5, c4 = fi & 31;
    async_copy_b128(hmat + (size_t)(row0 + r) * DD + c4 * 4,
                    As + r * APAD + c4 * 4);
  }
  async_wait0();
  __syncthreads();

  const int wave  = tid >> 5;
  const int lane  = tid & 31;
  const int lhalf = lane >> 4;          // 0: K even pair, 1: K odd pair
  const int l15   = lane & 15;
  const int n0    = wave * 16;

  v8f c = {};
  #pragma unroll
  for (int k0 = 0; k0 < DD; k0 += 4) {
    const int ka = k0 + 2 * lhalf;
    // A frag: lane -> row (l15), K = ka, ka+1  (contiguous -> ds_load_b64)
    v2f av = *(const v2f*)(As + l15 * APAD + ka);
    // B frag: W[ka][n0+l15], W[ka+1][n0+l15]
    v2f bv;
    bv.x = W[(size_t)ka * DD + n0 + l15];
    bv.y = W[(size_t)(ka + 1) * DD + n0 + l15];
    c = __builtin_amdgcn_wmma_f32_16x16x4_f32(false, av, false, bv,
                                              (short)0, c, false, false);
  }
  // C/D layout: VGPR r, lane l -> M = r + 8*(l>=16), Nn = l&15
  #pragma unroll
  for (int r = 0; r < 8; ++r) {
    const int row = row0 + r + 8 * lhalf;
    g[(size_t)row * DD + n0 + l15] = c[r];
  }
}

// =====================================================================
// Kernel 3: s_i / s_j row scores (tiny; 25600 dot products of length 128)
// =====================================================================
__global__ void gat_sij_kernel(const float* __restrict__ hmat,
                               const float* __restrict__ u1,
                               const float* __restrict__ u2,
                               float* __restrict__ s_i,
                               float* __restrict__ s_j) {
  const int idx = blockIdx.x * blockDim.x + threadIdx.x;  // (row, h)
  if (idx >= ROWS * HH) return;
  const int row = idx >> 3, h = idx & 7;
  float si = 0.f, sj = 0.f;
  #pragma unroll 8
  for (int d = 0; d < DD; ++d) {
    const float x = hmat[(size_t)row * DD + d];
    si += x * u1[d * HH + h];
    sj += x * u2[d * HH + h];
  }
  s_i[idx] = si;
  s_j[idx] = sj;
}

// =====================================================================
// Kernel 4 (bandwidth-critical): s_e[r][h] = dis_mat[r][:] . v[:][h]
// One wave per 16-row tile of dis_mat; 4 waves / block. A tiles are
// async-copied memory->LDS (no VGPR round-trip; ASYNCcnt pipeline);
// B fragments (loop-invariant per lane) preloaded into 64 VGPRs so the
// hot loop is ds_load_2addr_b64 + v_wmma only. Single pass over the
// 327.7 MB dis_mat -> ~14 us floor at 23.3 TB/s HBM.
// =====================================================================
__global__ void gat_se_kernel(const float* __restrict__ dis,
                              const float* __restrict__ vpad,
                              float* __restrict__ s_e) {
  __shared__ __align__(16) float Bs[DD * 16];       // 8 KB (v, padded N=16)
  __shared__ __align__(16) float As[4][16 * APAD];  // ~33 KB
  const int tid  = threadIdx.x;
  const int wave = tid >> 5;
  const int lane = tid & 31;

  // stage v (2048 floats / 128 thr)
  for (int fi = tid; fi < 512; fi += 128)
    ((float4*)Bs)[fi] = ((const float4*)vpad)[fi];

  // async-stage this wave's 16x128 dis tile: row r, 16B chunk = lane
  const size_t row0 = ((size_t)blockIdx.x * 4 + wave) * 16;
  const float* src  = dis + row0 * DD;
  float* dstA = &As[wave][0];
  #pragma unroll
  for (int r = 0; r < 16; ++r) {
    async_copy_b128(src + (size_t)r * DD + lane * 4,
                    dstA + r * APAD + lane * 4);
  }
  async_wait0();
  __syncthreads();   // Bs is cross-wave; async tile waited above

  const int lhalf = lane >> 4;
  const int l15   = lane & 15;

  // preload loop-invariant B fragments (v columns) into registers
  v2f bf[32];
  #pragma unroll
  for (int s = 0; s < 32; ++s) {
    const int ka = 4 * s + 2 * lhalf;
    bf[s].x = Bs[ka * 16 + l15];
    bf[s].y = Bs[(ka + 1) * 16 + l15];
  }

  v8f c = {};
  #pragma unroll
  for (int s = 0; s < 32; ++s) {
    const int ka = 4 * s + 2 * lhalf;
    v2f av = *(const v2f*)(dstA + l15 * APAD + ka);
    c = __builtin_amdgcn_wmma_f32_16x16x4_f32(false, av, false, bf[s],
                                              (short)0, c, false, false);
  }
  // only head columns 0..7 are real
  if (l15 < HH) {
    #pragma unroll
    for (int r = 0; r < 8; ++r) {
      const size_t row = row0 + r + 8 * lhalf;
      s_e[row * HH + l15] = c[r];
    }
  }
}

// =====================================================================
// Kernel 5: per-(b,i) masked leaky-relu + softmax over j + attn @ g.
// Block = 256 thr; thread t owns head h=t&7, rows j = (t>>3)+32k.
// Scores kept in LDS (200x8); two-pass softmax; output pass uses 128
// threads with fully coalesced g reads (g is L2/WGP$-resident).
// =====================================================================
__global__ void gat_attn_out_kernel(const float* __restrict__ s_e,
                                    const float* __restrict__ s_i,
                                    const float* __restrict__ s_j,
                                    const int*   __restrict__ adj,
                                    const float* __restrict__ g,
                                    float* __restrict__ out) {
  __shared__ float p[NN * HH];     // scores -> probabilities
  __shared__ float pm[32][HH];     // partial reductions
  __shared__ float mh[HH];         // per-head max
  __shared__ float sh[HH];         // per-head sum
  const int bi  = blockIdx.x;      // b*200 + i
  const int b   = bi / NN;
  const int tid = threadIdx.x;
  const int h   = tid & 7;
  const int j0  = tid >> 3;        // 0..31

  const float si = s_i[bi * HH + h];
  float m = -INFINITY;
  for (int j = j0; j < NN; j += 32) {
    float x = si + s_j[(b * NN + j) * HH + h]
                 + s_e[((size_t)bi * NN + j) * HH + h];
    x = (x > 0.f) ? x : NEG_SLOPE * x;                 // leaky_relu
    x = (adj[(size_t)bi * NN + j] == 0) ? -INFINITY : x;
    p[j * HH + h] = x;
    m = fmaxf(m, x);
  }
  pm[j0][h] = m;
  __syncthreads();
  if (tid < HH) {
    float mm = -INFINITY;
    for (int q = 0; q < 32; ++q) mm = fmaxf(mm, pm[q][tid]);
    mh[tid] = mm;
  }
  __syncthreads();

  const float mhh = mh[h];
  float s = 0.f;
  for (int j = j0; j < NN; j += 32) {
    const float ex = expf(p[j * HH + h] - mhh);
    p[j * HH + h] = ex;
    s += ex;
  }
  pm[j0][h] = s;
  __syncthreads();
  if (tid < HH) {
    float ss = 0.f;
    for (int q = 0; q < 32; ++q) ss += pm[q][tid];
    sh[tid] = ss;
  }
  __syncthreads();

  if (tid < 128) {                       // (head, feature) = (tid>>4, tid&15)
    const int h2 = tid >> 4, f = tid & 15;
    const float inv = 1.0f / sh[h2];
    float acc = 0.f;
    for (int j = 0; j < NN; ++j)
      acc += p[j * HH + h2] * g[((size_t)(b * NN + j)) * DD + h2 * NPH + f];
    out[(size_t)bi * DD + h2 * NPH + f] = acc * inv;
  }
}

// =====================================================================
extern "C" void kernel_launch(void* const* d_in, const int* in_sizes, int n_in,
                              void* d_out, int out_size, void* d_ws, size_t ws_size,
                              hipStream_t stream) {
  const float* hmat = (const float*)d_in[0];   // (16,200,128)
  const int*   adj  = (const int*)  d_in[1];   // (16,200,200,1)
  const float* dis  = (const float*)d_in[2];   // (16,200,200,128)
  const float* W    = (const float*)d_in[3];   // (128,128)
  const float* We   = (const float*)d_in[4];   // (128,128)
  const float* a    = (const float*)d_in[5];   // (48,)
  float* out = (float*)d_out;                  // (16,200,128)

  // workspace layout (floats): 22.3 MB total
  float* ws   = (float*)d_ws;
  float* u1   = ws;                            // 1024
  float* u2   = ws + 1024;                     // 1024
  float* vpad = ws + 2048;                     // 2048 (128x16)
  float* s_i  = ws + 4096;                     // 25600
  float* s_j  = ws + 4096 + 25600;             // 25600
  float* g    = ws + 4096 + 51200;             // 409600
  float* s_e  = ws + 4096 + 51200 + 409600;    // 5120000

  gat_prep_kernel<<<1, 128, 0, stream>>>(W, We, a, u1, u2, vpad);
  gat_gemm_g_kernel<<<ROWS / 16, 256, 0, stream>>>(hmat, W, g);
  gat_sij_kernel<<<(ROWS * HH + 255) / 256, 256, 0, stream>>>(hmat, u1, u2, s_i, s_j);
  gat_se_kernel<<<(int)(EROWS / 64), 128, 0, stream>>>(dis, vpad, s_e);
  gat_attn_out_kernel<<<ROWS, 256, 0, stream>>>(s_e, s_i, s_j, adj, g, out);
}